// CustomBCEWithLogitsLoss_61959198212241
// MI455X (gfx1250) — compile-verified
//
#include <hip/hip_runtime.h>
#include <hip/hip_bf16.h>
#include <math.h>

typedef float v2f __attribute__((ext_vector_type(2)));
typedef float v8f __attribute__((ext_vector_type(8)));

#define TPB  256
#define TOPK 20

// One block per row: streams the row once with B128 loads, computes the stable
// BCE partial sum and per-thread top-20 (top-20 of logits == top-20 of sigmoid
// probs), merges the sorted 20-lists in LDS, and emits per-row bce-sum and
// top20-loss-sum in deterministic order.
__global__ __launch_bounds__(TPB) void bce_topk_row_kernel(
    const float* __restrict__ logits,
    const float* __restrict__ targets,
    float* __restrict__ row_bce,
    float* __restrict__ row_top,
    int n)
{
    __shared__ float smv[TPB * TOPK];
    __shared__ int   smi[TPB * TOPK];
    __shared__ float wsum[16];
    __shared__ float tl[TOPK];

    const int row = blockIdx.x;
    const int tid = threadIdx.x;
    const float* __restrict__ lrow = logits  + (size_t)row * (size_t)n;
    const float* __restrict__ trow = targets + (size_t)row * (size_t)n;

    // per-thread top-20, kept ascending: a[0] = current min of the kept set
    float a[TOPK]; int ai[TOPK];
#pragma unroll
    for (int k = 0; k < TOPK; ++k) { a[k] = -INFINITY; ai[k] = 0; }

    // ---- streaming scan: float4 (global_load_b128) per thread per array
    const int nv = n >> 2;                       // n is a multiple of 4 (10000)
    const float4* __restrict__ l4 = (const float4*)lrow;
    const float4* __restrict__ t4 = (const float4*)trow;

    float s = 0.0f;  // BCE partial: max(x,0) - x*t + log1p(exp(-|x|))
    for (int c4 = tid; c4 < nv; c4 += TPB) {
        float4 xv = l4[c4];
        float4 tv = t4[c4];
        if (c4 + TPB < nv) {                     // gfx1250 global_prefetch_b8
            __builtin_prefetch((const float*)(l4 + c4 + TPB), 0, 0);
            __builtin_prefetch((const float*)(t4 + c4 + TPB), 0, 0);
        }
        const float xs[4] = {xv.x, xv.y, xv.z, xv.w};
        const float ts[4] = {tv.x, tv.y, tv.z, tv.w};
#pragma unroll
        for (int e = 0; e < 4; ++e) {
            const float x = xs[e];
            const float t = ts[e];
            const int col = (c4 << 2) + e;
            s += fmaxf(x, 0.0f) - x * t + log1pf(expf(-fabsf(x)));

            if (x > a[0]) {  // branch-free sorted insert, fully unrolled
#pragma unroll
                for (int j = 0; j < TOPK - 1; ++j) {
                    bool  shift = x > a[j + 1];
                    float keep  = (x > a[j]) ? x   : a[j];
                    int   keepi = (x > a[j]) ? col : ai[j];
                    a[j]  = shift ? a[j + 1]  : keep;
                    ai[j] = shift ? ai[j + 1] : keepi;
                }
                if (x > a[TOPK - 1]) { a[TOPK - 1] = x; ai[TOPK - 1] = col; }
            }
        }
    }

    // ---- wave-level reduction of s via V_WMMA_F32_16X16X4_F32 with B = ones.
    // A(16x4): lane i<16 holds {A[i][0],A[i][1]}={s,0}; lane i+16 holds {A[i][2],A[i][3]}.
    // => D[i][j] = s(lane i) + s(lane i+16) for every column j (B all-ones, layout-proof).
    // Uniform control flow here: EXEC is all ones as WMMA requires.
    v2f av; av[0] = s;    av[1] = 0.0f;
    v2f bv; bv[0] = 1.0f; bv[1] = 1.0f;
    v8f cacc = {};
    cacc = __builtin_amdgcn_wmma_f32_16x16x4_f32(
        /*neg_a=*/false, av, /*neg_b=*/false, bv,
        /*c_mod=*/(short)0, cacc, /*reuse_a=*/false, /*reuse_b=*/false);
    float tw = cacc[0] + cacc[1] + cacc[2] + cacc[3] +
               cacc[4] + cacc[5] + cacc[6] + cacc[7];
    // lanes 0..15 all hold sum(rows 0..7)  = sum of s over lanes {0..7,16..23};
    // lanes 16..31 hold sum(rows 8..15)    = sum of s over lanes {8..15,24..31}.
    const int lane = tid & 31;
    const int wid  = tid >> 5;
    if (lane == 0)  wsum[wid * 2 + 0] = tw;
    if (lane == 16) wsum[wid * 2 + 1] = tw;

    // stash per-thread top-20 (descending) for the LDS merge tree
#pragma unroll
    for (int k = 0; k < TOPK; ++k) {
        smv[tid * TOPK + k] = a[TOPK - 1 - k];
        smi[tid * TOPK + k] = ai[TOPK - 1 - k];
    }
    __syncthreads();

    if (tid == 0) {   // deterministic fixed-order block sum of the 16 half-wave sums
        float b = 0.0f;
#pragma unroll
        for (int i = 0; i < 16; ++i) b += wsum[i];
        row_bce[row] = b;
    }

    // merge tree: 256 sorted 20-lists -> 1 sorted 20-list (two-pointer top-20 merge)
    for (int stride = TPB / 2; stride >= 1; stride >>= 1) {
        if (tid < stride) {
            const int ab = tid * TOPK;
            const int bb = (tid + stride) * TOPK;
            float ov[TOPK]; int oi[TOPK];
            int p = 0, q = 0;
#pragma unroll
            for (int k = 0; k < TOPK; ++k) {
                float x0 = smv[ab + p];
                float x1 = smv[bb + q];
                bool  ta = (x0 >= x1);
                ov[k] = ta ? x0 : x1;
                oi[k] = ta ? smi[ab + p] : smi[bb + q];
                p += ta ? 1 : 0;
                q += ta ? 0 : 1;
            }
#pragma unroll
            for (int k = 0; k < TOPK; ++k) { smv[ab + k] = ov[k]; smi[ab + k] = oi[k]; }
        }
        __syncthreads();
    }

    // top-20 loss from winning logits: log p = min(x,0)-log1p(e^-|x|), clamp -100
    if (tid < TOPK) {
        float x   = smv[tid];
        int   col = smi[tid];
        col = (col >= 0 && col < n) ? col : 0;
        float t   = trow[col];
        float sp  = log1pf(expf(-fabsf(x)));
        float lp  = fminf(x,  0.0f) - sp;
        float l1p = fminf(-x, 0.0f) - sp;
        lp  = fmaxf(lp,  -100.0f);
        l1p = fmaxf(l1p, -100.0f);
        tl[tid] = -(t * lp + (1.0f - t) * l1p);
    }
    __syncthreads();
    if (tid == 0) {
        float r = 0.0f;
#pragma unroll
        for (int k = 0; k < TOPK; ++k) r += tl[k];
        row_top[row] = r;
    }
}

// Deterministic final reduction over rows + scalar combine.
__global__ __launch_bounds__(TPB) void finalize_kernel(
    const float* __restrict__ row_bce,
    const float* __restrict__ row_top,
    const float* __restrict__ bce_l,
    float* __restrict__ out,
    int B, int n)
{
    __shared__ float sb[TPB], st[TPB];
    const int tid = threadIdx.x;
    float b = 0.0f, t = 0.0f;
    for (int i = tid; i < B; i += TPB) { b += row_bce[i]; t += row_top[i]; }
    sb[tid] = b; st[tid] = t;
    __syncthreads();
    for (int s2 = TPB / 2; s2 > 0; s2 >>= 1) {
        if (tid < s2) { sb[tid] += sb[tid + s2]; st[tid] += st[tid + s2]; }
        __syncthreads();
    }
    if (tid == 0) {
        double bn = (double)B * (double)n;
        double bk = (double)B * (double)TOPK;
        out[0] = (float)((double)sb[0] / bn +
                         (double)bce_l[0] * ((double)st[0] / bk));
    }
}

extern "C" void kernel_launch(void* const* d_in, const int* in_sizes, int n_in,
                              void* d_out, int out_size, void* d_ws, size_t ws_size,
                              hipStream_t stream) {
    (void)n_in; (void)out_size; (void)ws_size;
    const float* logits  = (const float*)d_in[0];
    const float* targets = (const float*)d_in[1];
    const float* bce_l   = (const float*)d_in[2];
    float* out = (float*)d_out;

    const int N = 10000;                       // reference row length
    const int B = in_sizes[0] / N;             // 4096

    float* row_bce = (float*)d_ws;             // [B]
    float* row_top = row_bce + B;              // [B]

    bce_topk_row_kernel<<<B, TPB, 0, stream>>>(logits, targets, row_bce, row_top, N);
    finalize_kernel<<<1, TPB, 0, stream>>>(row_bce, row_top, bce_l, out, B, N);
}